// MultiLatentAttention_33200097198432
// MI455X (gfx1250) — compile-verified
//
#include <hip/hip_runtime.h>

// ---------------- problem constants (from reference) ----------------
#define S_LEN 2048
#define H_DIM 2048
#define NH 16
#define NKV 4
#define HD 128
#define RHD 64
#define QCOMP 1536
#define KVCOMP 512
#define EPSV 1e-6f
#define QKD (HD + RHD)                       // 192
#define SCALEV 0.07216878364870323f          // (HD+RHD)^-0.5 = 1/sqrt(192)

// ---------------- WMMA fragment / descriptor vector types ----------
typedef float v2f __attribute__((ext_vector_type(2)));
typedef float v8f __attribute__((ext_vector_type(8)));
typedef unsigned int v4u __attribute__((ext_vector_type(4)));
typedef int v4i __attribute__((ext_vector_type(4)));
typedef int v8i __attribute__((ext_vector_type(8)));

// ---------------- GEMM tiling -------------------------------------
// Block = 256 threads = 8 wave32s as 4(M) x 2(N); each wave owns a 32x32
// C tile (2x2 WMMA 16x16 tiles) -> block tile 128x64, K staged by 32.
#define BM 128
#define BN 64
#define BK 32
#define LDA_W (BK + 4)   // 36 floats: A-style LDS row (pad 4)
#define LDB_W (BN + 4)   // 68 floats: B-NN LDS row (pad 4)

#define HAVE_TDM __has_builtin(__builtin_amdgcn_tensor_load_to_lds)

#if __has_builtin(__builtin_amdgcn_s_wait_tensorcnt)
#define WAIT_TENSORCNT(n) __builtin_amdgcn_s_wait_tensorcnt(n)
#else
#define WAIT_TENSORCNT(n) asm volatile("s_wait_tensorcnt %0" ::"i"(n) : "memory")
#endif

#if HAVE_TDM
// Issue one TDM 2D tile load: tile_d0 contiguous f32 elements per row,
// tile_d1 rows, row stride `stride_elems` (f32), HW pads LDS rows by
// pad_amount+1 DWORDs every 2^(pad_interval+1) DWORDs (ISA 8.3-8.6).
__device__ __forceinline__ void tdm_load_2d(const void* gsrc, unsigned lds_addr,
                                            unsigned tile_d0, unsigned tile_d1,
                                            unsigned stride_elems,
                                            unsigned pad_interval,
                                            unsigned pad_amount) {
  unsigned long long ga = (unsigned long long)(size_t)gsrc;
  // group0: count=1 | lds_addr[63:32] ; global_addr[120:64] | type=2 [127:126]
  unsigned long long g0w0 = 1ull | ((unsigned long long)lds_addr << 32);
  unsigned long long g0w1 = (ga & 0x01FFFFFFFFFFFFFFull) | (2ull << 62);
  // group1: data_size=4B, pad_enable, pad geometry, dims/strides
  unsigned long long w0 = (2ull << 16) | (1ull << 20) |
                          ((unsigned long long)pad_interval << 22) |
                          ((unsigned long long)pad_amount << 25) |
                          ((unsigned long long)(tile_d0 & 0xFFFFu) << 48);
  unsigned long long w1 = 0ull |                       // tensor_dim0 hi = 0
                          ((unsigned long long)tile_d1 << 16) |  // tensor_dim1
                          ((unsigned long long)(tile_d0 & 0xFFFFu) << 48);
  unsigned long long st = stride_elems;
  unsigned long long w2 = ((unsigned long long)(tile_d1 & 0xFFFFu)) | (st << 32);
  unsigned long long w3 = (st >> 32) & 0xFFFFull;
  v4u g0;
  g0.x = (unsigned)g0w0; g0.y = (unsigned)(g0w0 >> 32);
  g0.z = (unsigned)g0w1; g0.w = (unsigned)(g0w1 >> 32);
  v8i g1;
  g1[0] = (int)w0; g1[1] = (int)(w0 >> 32);
  g1[2] = (int)w1; g1[3] = (int)(w1 >> 32);
  g1[4] = (int)w2; g1[5] = (int)(w2 >> 32);
  g1[6] = (int)w3; g1[7] = (int)(w3 >> 32);
  v4i z = {0, 0, 0, 0};
#if defined(__clang_major__) && __clang_major__ >= 23
  v8i z8 = {0, 0, 0, 0, 0, 0, 0, 0};
  __builtin_amdgcn_tensor_load_to_lds(g0, g1, z, z, z8, 0);
#else
  __builtin_amdgcn_tensor_load_to_lds(g0, g1, z, z, 0);
#endif
}
#endif  // HAVE_TDM

// ---------------- fallback synchronous tile loaders ----------------
__device__ __forceinline__ void sload_A(float (*dst)[LDA_W],
                                        const float* __restrict__ src, int ld,
                                        int tid) {
#pragma unroll
  for (int it = 0; it < 4; ++it) {  // 128x32 floats, 4 float4 / thread
    int idx = tid + it * 256;
    int r = idx >> 3, cv = (idx & 7) << 2;
    float4 v = *reinterpret_cast<const float4*>(src + (size_t)r * ld + cv);
    *reinterpret_cast<float4*>(&dst[r][cv]) = v;
  }
}
__device__ __forceinline__ void sload_Bnn(float (*dst)[LDB_W],
                                          const float* __restrict__ src, int ld,
                                          int tid) {
#pragma unroll
  for (int it = 0; it < 2; ++it) {  // 32x64 floats, 2 float4 / thread
    int idx = tid + it * 256;
    int r = idx >> 4, cv = (idx & 15) << 2;
    float4 v = *reinterpret_cast<const float4*>(src + (size_t)r * ld + cv);
    *reinterpret_cast<float4*>(&dst[r][cv]) = v;
  }
}
__device__ __forceinline__ void sload_Bt(float (*dst)[LDA_W],
                                         const float* __restrict__ src, int ld,
                                         int tid) {
#pragma unroll
  for (int it = 0; it < 2; ++it) {  // 64x32 floats, 2 float4 / thread
    int idx = tid + it * 256;
    int r = idx >> 3, cv = (idx & 7) << 2;
    float4 v = *reinterpret_cast<const float4*>(src + (size_t)r * ld + cv);
    *reinterpret_cast<float4*>(&dst[r][cv]) = v;
  }
}

// ---------------- one K-stage of WMMA (2x2 tiles per wave) ---------
// f32 16x16x4 fragment layout (ISA 7.12.2):
//   A: lanes 0-15 -> (M=lane, K=k,k+1), lanes 16-31 -> (M=lane-16, K=k+2,k+3)
//   B: K striped over VGPRs, N over lanes (same lane split for K pairs)
__device__ __forceinline__ void mma_stage_nn(const float (*As)[LDA_W],
                                             const float (*Bs)[LDB_W],
                                             int waveM, int waveN, int l16,
                                             int hi, v8f acc[4]) {
  const int mr = waveM * 32 + l16;
  const int nc = waveN * 32 + l16;
#pragma unroll
  for (int kk = 0; kk < BK; kk += 4) {
    int ka = kk + 2 * hi;
    v2f a0, a1, b0, b1;
    a0.x = As[mr][ka];      a0.y = As[mr][ka + 1];
    a1.x = As[mr + 16][ka]; a1.y = As[mr + 16][ka + 1];
    b0.x = Bs[ka][nc];      b0.y = Bs[ka + 1][nc];
    b1.x = Bs[ka][nc + 16]; b1.y = Bs[ka + 1][nc + 16];
    acc[0] = __builtin_amdgcn_wmma_f32_16x16x4_f32(false, a0, false, b0, (short)0, acc[0], false, false);
    acc[1] = __builtin_amdgcn_wmma_f32_16x16x4_f32(false, a0, false, b1, (short)0, acc[1], false, false);
    acc[2] = __builtin_amdgcn_wmma_f32_16x16x4_f32(false, a1, false, b0, (short)0, acc[2], false, false);
    acc[3] = __builtin_amdgcn_wmma_f32_16x16x4_f32(false, a1, false, b1, (short)0, acc[3], false, false);
  }
}
__device__ __forceinline__ void mma_stage_nt(const float (*As)[LDA_W],
                                             const float (*Bt)[LDA_W],
                                             int waveM, int waveN, int l16,
                                             int hi, v8f acc[4]) {
  const int mr = waveM * 32 + l16;
  const int nr = waveN * 32 + l16;
#pragma unroll
  for (int kk = 0; kk < BK; kk += 4) {
    int ka = kk + 2 * hi;
    v2f a0, a1, b0, b1;
    a0.x = As[mr][ka];      a0.y = As[mr][ka + 1];
    a1.x = As[mr + 16][ka]; a1.y = As[mr + 16][ka + 1];
    b0.x = Bt[nr][ka];      b0.y = Bt[nr][ka + 1];
    b1.x = Bt[nr + 16][ka]; b1.y = Bt[nr + 16][ka + 1];
    acc[0] = __builtin_amdgcn_wmma_f32_16x16x4_f32(false, a0, false, b0, (short)0, acc[0], false, false);
    acc[1] = __builtin_amdgcn_wmma_f32_16x16x4_f32(false, a0, false, b1, (short)0, acc[1], false, false);
    acc[2] = __builtin_amdgcn_wmma_f32_16x16x4_f32(false, a1, false, b0, (short)0, acc[2], false, false);
    acc[3] = __builtin_amdgcn_wmma_f32_16x16x4_f32(false, a1, false, b1, (short)0, acc[3], false, false);
  }
}

// ---------------- generic C = A(MxK) * B(KxN), row-major -----------
__global__ __launch_bounds__(256) void wmma_gemm_nn(
    const float* __restrict__ A, int lda, const float* __restrict__ B, int ldb,
    float* __restrict__ C, int ldc, int K) {
  __shared__ __align__(16) float As[2][BM][LDA_W];
  __shared__ __align__(16) float Bs[2][BK][LDB_W];
  const int tid = threadIdx.x;
  const int lane = tid & 31, wave = tid >> 5;
  const int waveM = wave & 3, waveN = wave >> 2;
  const int l16 = lane & 15, hi = lane >> 4;
  const size_t m0 = (size_t)blockIdx.y * BM;
  const size_t n0 = (size_t)blockIdx.x * BN;
  const int nst = K / BK;
  v8f acc[4] = {};

#if HAVE_TDM
  if (wave == 0) {
    tdm_load_2d(A + m0 * lda, (unsigned)(size_t)&As[0][0][0], BK, BM, lda, 4, 3);
    tdm_load_2d(B + n0, (unsigned)(size_t)&Bs[0][0][0], BN, BK, ldb, 5, 3);
  }
  for (int i = 0; i < nst; ++i) {
    const int cur = i & 1;
    if (wave == 0) {
      if (i + 1 < nst) {
        tdm_load_2d(A + m0 * lda + (size_t)(i + 1) * BK,
                    (unsigned)(size_t)&As[cur ^ 1][0][0], BK, BM, lda, 4, 3);
        tdm_load_2d(B + (size_t)(i + 1) * BK * ldb + n0,
                    (unsigned)(size_t)&Bs[cur ^ 1][0][0], BN, BK, ldb, 5, 3);
        WAIT_TENSORCNT(2);  // stage i complete; stage i+1 in flight
      } else {
        WAIT_TENSORCNT(0);
      }
    }
    __syncthreads();
    mma_stage_nn(As[cur], Bs[cur], waveM, waveN, l16, hi, acc);
    __syncthreads();
  }
#else
  sload_A(As[0], A + m0 * lda, lda, tid);
  sload_Bnn(Bs[0], B + n0, ldb, tid);
  for (int i = 0; i < nst; ++i) {
    const int cur = i & 1;
    __syncthreads();
    if (i + 1 < nst) {
      sload_A(As[cur ^ 1], A + m0 * lda + (size_t)(i + 1) * BK, lda, tid);
      sload_Bnn(Bs[cur ^ 1], B + (size_t)(i + 1) * BK * ldb + n0, ldb, tid);
    }
    mma_stage_nn(As[cur], Bs[cur], waveM, waveN, l16, hi, acc);
  }
#endif

  const size_t cm0 = m0 + waveM * 32, cn0 = n0 + waveN * 32;
#pragma unroll
  for (int t = 0; t < 4; ++t) {
    size_t rb = cm0 + ((t >> 1) << 4) + hi * 8;
    size_t cc = cn0 + ((t & 1) << 4) + l16;
#pragma unroll
    for (int i = 0; i < 8; ++i) C[(rb + i) * ldc + cc] = acc[t][i];
  }
}

// ---------------- scores = (Q @ K^T) * SCALE with causal mask ------
__global__ __launch_bounds__(256) void wmma_scores(
    const float* __restrict__ Q, const float* __restrict__ Kmat,
    float* __restrict__ attn) {
  __shared__ __align__(16) float As[2][BM][LDA_W];
  __shared__ __align__(16) float Bt[2][BN][LDA_W];
  const int h = blockIdx.z;
  const float* A = Q + (size_t)h * S_LEN * QKD;
  const float* B = Kmat + (size_t)(h >> 2) * S_LEN * QKD;
  float* Ch = attn + (size_t)h * S_LEN * S_LEN;
  const int tid = threadIdx.x;
  const int lane = tid & 31, wave = tid >> 5;
  const int waveM = wave & 3, waveN = wave >> 2;
  const int l16 = lane & 15, hi = lane >> 4;
  const size_t m0 = (size_t)blockIdx.y * BM;
  const size_t n0 = (size_t)blockIdx.x * BN;
  const int nst = QKD / BK;  // 6
  v8f acc[4] = {};

#if HAVE_TDM
  if (wave == 0) {
    tdm_load_2d(A + m0 * QKD, (unsigned)(size_t)&As[0][0][0], BK, BM, QKD, 4, 3);
    tdm_load_2d(B + n0 * QKD, (unsigned)(size_t)&Bt[0][0][0], BK, BN, QKD, 4, 3);
  }
  for (int i = 0; i < nst; ++i) {
    const int cur = i & 1;
    if (wave == 0) {
      if (i + 1 < nst) {
        tdm_load_2d(A + m0 * QKD + (size_t)(i + 1) * BK,
                    (unsigned)(size_t)&As[cur ^ 1][0][0], BK, BM, QKD, 4, 3);
        tdm_load_2d(B + n0 * QKD + (size_t)(i + 1) * BK,
                    (unsigned)(size_t)&Bt[cur ^ 1][0][0], BK, BN, QKD, 4, 3);
        WAIT_TENSORCNT(2);
      } else {
        WAIT_TENSORCNT(0);
      }
    }
    __syncthreads();
    mma_stage_nt(As[cur], Bt[cur], waveM, waveN, l16, hi, acc);
    __syncthreads();
  }
#else
  sload_A(As[0], A + m0 * QKD, QKD, tid);
  sload_Bt(Bt[0], B + n0 * QKD, QKD, tid);
  for (int i = 0; i < nst; ++i) {
    const int cur = i & 1;
    __syncthreads();
    if (i + 1 < nst) {
      sload_A(As[cur ^ 1], A + m0 * QKD + (size_t)(i + 1) * BK, QKD, tid);
      sload_Bt(Bt[cur ^ 1], B + n0 * QKD + (size_t)(i + 1) * BK, QKD, tid);
    }
    mma_stage_nt(As[cur], Bt[cur], waveM, waveN, l16, hi, acc);
  }
#endif

  const size_t cm0 = m0 + waveM * 32, cn0 = n0 + waveN * 32;
#pragma unroll
  for (int t = 0; t < 4; ++t) {
    size_t rb = cm0 + ((t >> 1) << 4) + hi * 8;
    size_t cc = cn0 + ((t & 1) << 4) + l16;
#pragma unroll
    for (int i = 0; i < 8; ++i) {
      size_t row = rb + i;
      float v = acc[t][i] * SCALEV;
      Ch[row * S_LEN + cc] = (cc <= row) ? v : -1.0e9f;
    }
  }
}

// ---------------- RMS norm (in place, one block per row) -----------
__global__ __launch_bounds__(256) void rmsnorm_kernel(float* __restrict__ Y,
                                                      const float* __restrict__ w,
                                                      int n) {
  float* y = Y + (size_t)blockIdx.x * n;
  float ss = 0.f;
  for (int c = threadIdx.x; c < n; c += 256) {
    float t = y[c];
    ss += t * t;
  }
  __shared__ float red[8];
  __shared__ float scale_s;
  for (int o = 16; o > 0; o >>= 1) ss += __shfl_down(ss, o, 32);
  if ((threadIdx.x & 31) == 0) red[threadIdx.x >> 5] = ss;
  __syncthreads();
  if (threadIdx.x == 0) {
    float tot = 0.f;
    for (int i = 0; i < 8; ++i) tot += red[i];
    scale_s = rsqrtf(tot / (float)n + EPSV);
  }
  __syncthreads();
  const float sc = scale_s;
  for (int c = threadIdx.x; c < n; c += 256) y[c] = y[c] * sc * w[c];
}

// ---------------- RoPE + head assembly -----------------------------
__global__ __launch_bounds__(256) void rope_assemble(
    const float* __restrict__ qc, const float* __restrict__ qr,
    const float* __restrict__ kc, const float* __restrict__ kr,
    const float* __restrict__ cosb, const float* __restrict__ sinb,
    float* __restrict__ qbuf, float* __restrict__ kbuf) {
  const int s = blockIdx.x;
  const float* cs = cosb + (size_t)s * RHD;
  const float* sn = sinb + (size_t)s * RHD;
  for (int idx = threadIdx.x; idx < NH * QKD; idx += 256) {
    int h = idx / QKD, d = idx % QKD;
    float val;
    if (d < HD) {
      val = qc[(size_t)s * (NH * HD) + h * HD + d];
    } else {
      int j = d - HD;
      const float* qh = qr + (size_t)s * (NH * RHD) + h * RHD;
      float x = qh[j];
      float other = (j < RHD / 2) ? -qh[j + RHD / 2] : qh[j - RHD / 2];
      val = x * cs[j] + other * sn[j];
    }
    qbuf[(size_t)h * S_LEN * QKD + (size_t)s * QKD + d] = val;
  }
  for (int idx = threadIdx.x; idx < NKV * QKD; idx += 256) {
    int h = idx / QKD, d = idx % QKD;
    float val;
    if (d < HD) {
      val = kc[(size_t)s * (NKV * HD) + h * HD + d];
    } else {
      int j = d - HD;
      const float* kh = kr + (size_t)s * RHD;
      float x = kh[j];
      float other = (j < RHD / 2) ? -kh[j + RHD / 2] : kh[j - RHD / 2];
      val = x * cs[j] + other * sn[j];
    }
    kbuf[(size_t)h * S_LEN * QKD + (size_t)s * QKD + d] = val;
  }
}

// ---------------- row softmax (in place), one block per row --------
__global__ __launch_bounds__(256) void softmax_kernel(float* __restrict__ attn) {
  float* p = attn + (size_t)blockIdx.x * S_LEN;
  __shared__ float red[8];
  __shared__ float bval;
  const int lane = threadIdx.x & 31, wid = threadIdx.x >> 5;
  float m = -1e30f;
  for (int c = threadIdx.x; c < S_LEN; c += 256) m = fmaxf(m, p[c]);
  for (int o = 16; o > 0; o >>= 1) m = fmaxf(m, __shfl_down(m, o, 32));
  if (lane == 0) red[wid] = m;
  __syncthreads();
  if (threadIdx.x == 0) {
    float t = red[0];
    for (int i = 1; i < 8; ++i) t = fmaxf(t, red[i]);
    bval = t;
  }
  __syncthreads();
  m = bval;
  float ssum = 0.f;
  for (int c = threadIdx.x; c < S_LEN; c += 256) ssum += __expf(p[c] - m);
  for (int o = 16; o > 0; o >>= 1) ssum += __shfl_down(ssum, o, 32);
  if (lane == 0) red[wid] = ssum;
  __syncthreads();
  if (threadIdx.x == 0) {
    float t = 0.f;
    for (int i = 0; i < 8; ++i) t += red[i];
    bval = t;
  }
  __syncthreads();
  const float inv = 1.f / bval;
  for (int c = threadIdx.x; c < S_LEN; c += 256) p[c] = __expf(p[c] - m) * inv;
}

// ---------------- host orchestration -------------------------------
extern "C" void kernel_launch(void* const* d_in, const int* in_sizes, int n_in,
                              void* d_out, int out_size, void* d_ws,
                              size_t ws_size, hipStream_t stream) {
  (void)in_sizes; (void)n_in; (void)out_size; (void)ws_size;
  const float* X        = (const float*)d_in[0];
  const float* cosb     = (const float*)d_in[1];
  const float* sinb     = (const float*)d_in[2];
  /* d_in[3] = mask: causal mask computed analytically in wmma_scores */
  const float* w_down_q  = (const float*)d_in[4];
  const float* w_up_q    = (const float*)d_in[5];
  const float* w_qr      = (const float*)d_in[6];
  const float* w_down_kv = (const float*)d_in[7];
  const float* w_up_k    = (const float*)d_in[8];
  const float* w_up_v    = (const float*)d_in[9];
  const float* w_kr      = (const float*)d_in[10];
  const float* w_o       = (const float*)d_in[11];
  const float* qnw       = (const float*)d_in[12];
  const float* knw       = (const float*)d_in[13];

  float* out  = (float*)d_out;                 // [S, H]
  float* attn = out + (size_t)S_LEN * H_DIM;   // [NH, S, S]

  float* ws  = (float*)d_ws;
  float* cq  = ws;  ws += (size_t)S_LEN * QCOMP;
  float* ckv = ws;  ws += (size_t)S_LEN * KVCOMP;
  float* krb = ws;  ws += (size_t)S_LEN * RHD;
  float* qcb = ws;  ws += (size_t)S_LEN * NH * HD;
  float* qrb = ws;  ws += (size_t)S_LEN * NH * RHD;
  float* kcb = ws;  ws += (size_t)S_LEN * NKV * HD;
  float* vb  = ws;  ws += (size_t)S_LEN * NKV * HD;
  float* qb  = ws;  ws += (size_t)NH * S_LEN * QKD;
  float* kb  = ws;  ws += (size_t)NKV * S_LEN * QKD;
  float* ctx = ws;  ws += (size_t)S_LEN * NH * HD;

  const dim3 blk(256);
  auto gemm = [&](const float* A, int lda, const float* B, int ldb, float* C,
                  int ldc, int M, int N, int K) {
    dim3 grid(N / BN, M / BM);
    wmma_gemm_nn<<<grid, blk, 0, stream>>>(A, lda, B, ldb, C, ldc, K);
  };

  // Down projections
  gemm(X, H_DIM, w_down_q, QCOMP, cq, QCOMP, S_LEN, QCOMP, H_DIM);
  gemm(X, H_DIM, w_down_kv, KVCOMP, ckv, KVCOMP, S_LEN, KVCOMP, H_DIM);
  gemm(X, H_DIM, w_kr, RHD, krb, RHD, S_LEN, RHD, H_DIM);

  // RMS norms
  rmsnorm_kernel<<<S_LEN, blk, 0, stream>>>(cq, qnw, QCOMP);
  rmsnorm_kernel<<<S_LEN, blk, 0, stream>>>(ckv, knw, KVCOMP);

  // Up projections
  gemm(cq, QCOMP, w_up_q, NH * HD, qcb, NH * HD, S_LEN, NH * HD, QCOMP);
  gemm(cq, QCOMP, w_qr, NH * RHD, qrb, NH * RHD, S_LEN, NH * RHD, QCOMP);
  gemm(ckv, KVCOMP, w_up_k, NKV * HD, kcb, NKV * HD, S_LEN, NKV * HD, KVCOMP);
  gemm(ckv, KVCOMP, w_up_v, NKV * HD, vb, NKV * HD, S_LEN, NKV * HD, KVCOMP);

  // RoPE + assemble per-head Q/K
  rope_assemble<<<S_LEN, blk, 0, stream>>>(qcb, qrb, kcb, krb, cosb, sinb, qb, kb);

  // Attention scores (scale + causal mask fused)
  {
    dim3 grid(S_LEN / BN, S_LEN / BM, NH);
    wmma_scores<<<grid, blk, 0, stream>>>(qb, kb, attn);
  }

  // Softmax rows; probabilities written to d_out attn region
  softmax_kernel<<<NH * S_LEN, blk, 0, stream>>>(attn);

  // ctx[h] = attn[h] @ V[h/4]
  for (int h = 0; h < NH; ++h) {
    gemm(attn + (size_t)h * S_LEN * S_LEN, S_LEN,
         vb + (size_t)(h >> 2) * HD, NKV * HD,
         ctx + (size_t)h * HD, NH * HD, S_LEN, HD, S_LEN);
  }

  // Output projection
  gemm(ctx, NH * HD, w_o, H_DIM, out, H_DIM, S_LEN, H_DIM, NH * HD);
}